// SuperPointMatching_OT_2370821948165
// MI455X (gfx1250) — compile-verified
//
#include <hip/hip_runtime.h>
#include <hip/hip_bf16.h>

#define NM    8193      // matrix dim incl. dustbin
#define NI    8192      // inner dim
#define KCORR 2048
#define MAXC  8192
#define NT    128       // threshold ladder length (loop provably stops < ~52)

typedef float v2f __attribute__((ext_vector_type(2)));
typedef float v8f __attribute__((ext_vector_type(8)));

// ---------------------------------------------------------------- zero scratch
__global__ void k_zero(unsigned* __restrict__ ghist) {
    int i = threadIdx.x;
    if (i < NT) ghist[i] = 0u;
}

// ---------------------------------------------------------------- pass 1: per-row + global histogram over threshold ladder
__global__ void k_hist(const float* __restrict__ ms, const float* __restrict__ refw,
                       const float* __restrict__ srcw, unsigned* __restrict__ rowHist,
                       unsigned* __restrict__ ghist) {
    __shared__ float    t[NT];
    __shared__ unsigned h[NT + 1];
    const int tid = threadIdx.x;
    const int row = blockIdx.x;

    if (tid == 0) {                       // exact f32 replica of reference ladder
        float x = 0.5f;
        for (int k = 0; k < NT; ++k) { t[k] = x; x -= 0.01f; }
    }
    for (int k = tid; k < NT + 1; k += blockDim.x) h[k] = 0u;
    __syncthreads();

    const float rw = refw[row];
    const float* __restrict__ srow = ms + (size_t)row * NM;

    for (int j = tid; j < NI; j += blockDim.x) {
        float sc = srow[j];
        float v  = expf(sc) * (rw * srcw[j]);   // match reference op order
        int b;
        if (!(v > t[NT - 1])) {
            b = NT;                              // below the whole ladder: discard
        } else {                                 // smallest k with v > t[k]
            int lo = 0, hi = NT - 1;
            while (lo < hi) {
                int mid = (lo + hi) >> 1;
                if (v > t[mid]) hi = mid; else lo = mid + 1;
            }
            b = lo;
        }
        if (b < NT) atomicAdd(&h[b], 1u);
    }
    __syncthreads();

    for (int k = tid; k < NT; k += blockDim.x) {
        unsigned c = h[k];
        rowHist[(size_t)row * NT + k] = c;
        if (c) atomicAdd(&ghist[k], c);
    }
}

// ---------------------------------------------------------------- find final threshold (replicates while_loop)
__global__ void k_findthres(const unsigned* __restrict__ ghist,
                            int* __restrict__ kf_out, float* __restrict__ thres_out) {
    if (threadIdx.x == 0) {
        float x = 0.5f;
        unsigned c = 0;
        int kf = NT - 1;
        float th = x;
        bool found = false;
        for (int k = 0; k < NT; ++k) {
            c += ghist[k];
            if (c >= (unsigned)KCORR) { kf = k; th = x; found = true; break; }
            x -= 0.01f;
        }
        if (!found) { kf = NT - 1; th = x; }
        *kf_out = kf;
        *thres_out = th;
    }
}

// ---------------------------------------------------------------- rowCount = rowHist(8192x128) x selector(128) via WMMA
// A: 16x4 f32 tile (lanes 0-15: M=0..15,K=0,1 ; lanes 16-31: M=0..15,K=2,3)
// B: selector broadcast to every N column -> exact N mapping irrelevant;
//    consistent K mapping of A and B makes the contraction correct.
__global__ void k_rowcount(const unsigned* __restrict__ rowHist,
                           const int* __restrict__ kf_p,
                           unsigned* __restrict__ rowCounts) {
    const int lane    = threadIdx.x;          // 32 threads = 1 wave
    const int rowbase = blockIdx.x * 16;
    const int m       = lane & 15;
    const int khalf   = (lane >> 4) << 1;     // 0 or 2
    const int kf      = *kf_p;

    const unsigned* __restrict__ hrow = rowHist + (size_t)(rowbase + m) * NT;
    v8f acc = {};
    for (int c = 0; c < NT / 4; ++c) {
        int k0 = c * 4 + khalf;
        v2f a, b;
        a.x = (float)hrow[k0];
        a.y = (float)hrow[k0 + 1];
        b.x = (k0     <= kf) ? 1.0f : 0.0f;
        b.y = (k0 + 1 <= kf) ? 1.0f : 0.0f;
        acc = __builtin_amdgcn_wmma_f32_16x16x4_f32(
                  false, a, false, b, (short)0, acc, false, false);
    }
    // D layout: vgpr r, lanes<16 -> M=r ; lanes>=16 -> M=8+r ; N=lane&15
    if ((lane & 15) == 0) {
        int mb = (lane >> 4) * 8;
        for (int r = 0; r < 8; ++r)
            rowCounts[rowbase + mb + r] = (unsigned)(acc[r] + 0.5f);
    }
}

// ---------------------------------------------------------------- exclusive scan of 8192 row counts (1 block)
__global__ void k_scan(const unsigned* __restrict__ rowCounts,
                       unsigned* __restrict__ rowOffsets) {
    __shared__ unsigned s[1024];
    const int tid = threadIdx.x;              // 1024 threads x 8 rows
    unsigned loc[8];
    unsigned sum = 0;
    for (int i = 0; i < 8; ++i) { loc[i] = sum; sum += rowCounts[tid * 8 + i]; }
    s[tid] = sum;
    __syncthreads();
    for (int off = 1; off < 1024; off <<= 1) {
        unsigned v = (tid >= off) ? s[tid - off] : 0u;
        __syncthreads();
        s[tid] += v;
        __syncthreads();
    }
    unsigned base = s[tid] - sum;             // exclusive base of this thread
    for (int i = 0; i < 8; ++i) rowOffsets[tid * 8 + i] = base + loc[i];
}

// ---------------------------------------------------------------- default-fill output (d_out is poisoned)
__global__ void k_init(float* __restrict__ out) {
    int i = blockIdx.x * blockDim.x + threadIdx.x;
    if (i < MAXC) {
        out[i]            = -1.0f;   // ref indices
        out[MAXC + i]     = -1.0f;   // src indices
        out[2 * MAXC + i] = 0.0f;    // scores
    }
}

// ---------------------------------------------------------------- pass 2: ordered row-major compaction (wave32 ballot)
__global__ void k_emit(const float* __restrict__ ms, const float* __restrict__ refw,
                       const float* __restrict__ srcw,
                       const unsigned* __restrict__ rowOffsets,
                       const float* __restrict__ thres_p, float* __restrict__ out) {
    __shared__ unsigned wsum[8];
    const int tid = threadIdx.x;              // 256
    const int row = blockIdx.x;
    const unsigned base = rowOffsets[row];
    if (base >= MAXC) return;                 // uniform: whole row beyond cap

    const float thres = *thres_p;
    const float rw    = refw[row];
    const float* __restrict__ srow = ms + (size_t)row * NM;
    const int lane = tid & 31, wid = tid >> 5;

    unsigned running = 0;
    for (int c0 = 0; c0 < NI; c0 += 256) {
        int j = c0 + tid;
        float v = expf(srow[j]) * (rw * srcw[j]);   // identical FP ops to pass 1
        bool pred = v > thres;
        unsigned bal = __builtin_amdgcn_ballot_w32(pred);
        if (lane == 0) wsum[wid] = (unsigned)__popc(bal);
        __syncthreads();
        unsigned wpre = 0, tot = 0;
        for (int w = 0; w < 8; ++w) {
            unsigned cw = wsum[w];
            if (w < wid) wpre += cw;
            tot += cw;
        }
        if (pred) {
            unsigned p = base + running + wpre +
                         (unsigned)__popc(bal & ((1u << lane) - 1u));
            if (p < MAXC) {
                out[p]            = (float)row;
                out[MAXC + p]     = (float)j;
                out[2 * MAXC + p] = v;
            }
        }
        running += tot;
        if (base + running >= MAXC) break;    // uniform early exit
        __syncthreads();
    }
}

// ---------------------------------------------------------------- launcher
extern "C" void kernel_launch(void* const* d_in, const int* in_sizes, int n_in,
                              void* d_out, int out_size, void* d_ws, size_t ws_size,
                              hipStream_t stream) {
    (void)in_sizes; (void)n_in; (void)out_size; (void)ws_size;
    const float* ms   = (const float*)d_in[0];   // (8193, 8193) f32
    const float* refw = (const float*)d_in[1];   // (8192,) f32
    const float* srcw = (const float*)d_in[2];   // (8192,) f32
    float* out = (float*)d_out;                  // 3 * 8192 f32

    char* ws = (char*)d_ws;
    unsigned* ghist      = (unsigned*)(ws);                       // 128 u32
    int*      kf_p       = (int*)     (ws + 512);
    float*    thres_p    = (float*)   (ws + 516);
    unsigned* rowCounts  = (unsigned*)(ws + 4096);                // 8192 u32
    unsigned* rowOffsets = (unsigned*)(ws + 4096 + NI * 4);       // 8192 u32
    unsigned* rowHist    = (unsigned*)(ws + 4096 + 2 * NI * 4);   // 8192*128 u32 (4 MB)

    k_zero     <<<1,        128,  0, stream>>>(ghist);
    k_hist     <<<NI,       256,  0, stream>>>(ms, refw, srcw, rowHist, ghist);
    k_findthres<<<1,        32,   0, stream>>>(ghist, kf_p, thres_p);
    k_rowcount <<<NI / 16,  32,   0, stream>>>(rowHist, kf_p, rowCounts);
    k_scan     <<<1,        1024, 0, stream>>>(rowCounts, rowOffsets);
    k_init     <<<MAXC/256, 256,  0, stream>>>(out);
    k_emit     <<<NI,       256,  0, stream>>>(ms, refw, srcw, rowOffsets, thres_p, out);
}